// PowerMethodModel_3083786518609
// MI455X (gfx1250) — compile-verified
//
#include <hip/hip_runtime.h>
#include <math.h>

// Problem dims (fixed by the reference).
#define BB 8
#define HH 256
#define WW 256
#define ZZ 64
#define NTOT (BB*HH*WW*ZZ)      // 33554432
#define TW 64                   // w-tile per block
#define NCOL 66                 // TW + 2 halo columns
#define SLAB (NCOL*ZZ)          // 4224 floats per h-slab
#define SLAB_B (SLAB*4)         // 16896 bytes
#define CHUNKS (3*NCOL*16)      // 3168 16-byte chunks per block (3 slabs)
#define NBLK (BB*HH*(WW/TW))    // 8192 stencil blocks
#define PPB (HH*(WW/TW))        // 1024 partials per batch

typedef float v2f __attribute__((ext_vector_type(2)));
typedef float v8f __attribute__((ext_vector_type(8)));

// ---- CDNA5 async global->LDS staging (ASYNCcnt path) ------------------------
__device__ __forceinline__ void async_load_b128(unsigned lds_byte_off,
                                                unsigned long long gaddr) {
  asm volatile("global_load_async_to_lds_b128 %0, %1, off"
               :: "v"(lds_byte_off), "v"(gaddr) : "memory");
}
__device__ __forceinline__ void wait_async0() {
  asm volatile("s_wait_asynccnt 0" ::: "memory");
}

// ---- CDNA5 WMMA-based wave reduction ---------------------------------------
// V_WMMA_F32_16X16X4_F32 with B = ones: D[m][n] = sum_k A[m][k].
// A layout (16x4 f32): lanes 0-15 -> (M=lane, K=0/1), lanes 16-31 -> (M=lane-16, K=2/3),
// so with a = {v, 0}: row m = {v_m, 0, v_{m+16}, 0}.  D component j on lanes 0-15 is
// v_j + v_{j+16}; on lanes 16-31 it is v_{j+8} + v_{j+24}.  Summing the 8 D
// components then adding the value from lane^16 yields the full 32-lane sum.
__device__ __forceinline__ float waveRedSum(float v) {
  v2f a; a[0] = v;   a[1] = 0.f;
  v2f b; b[0] = 1.f; b[1] = 1.f;
  v8f c = {0.f, 0.f, 0.f, 0.f, 0.f, 0.f, 0.f, 0.f};
  v8f d = __builtin_amdgcn_wmma_f32_16x16x4_f32(false, a, false, b,
                                                (short)0, c, false, false);
  float s = d[0] + d[1] + d[2] + d[3] + d[4] + d[5] + d[6] + d[7];
  s += __shfl_xor(s, 16, 32);
  return s;                    // wave total in every lane
}

// Deterministic block reductions (fixed dataflow, 8 waves per 256-thread block).
__device__ __forceinline__ float blockSum(float v, float* red8) {
  float ws = waveRedSum(v);
  int t = threadIdx.x;
  if ((t & 31) == 0) red8[t >> 5] = ws;
  __syncthreads();
  float tot = red8[0] + red8[1] + red8[2] + red8[3] +
              red8[4] + red8[5] + red8[6] + red8[7];
  __syncthreads();
  return tot;
}
__device__ __forceinline__ float blockMax(float v, float* red8) {
  float s = v;
  for (int off = 16; off > 0; off >>= 1) s = fmaxf(s, __shfl_xor(s, off, 32));
  int t = threadIdx.x;
  if ((t & 31) == 0) red8[t >> 5] = s;
  __syncthreads();
  float m = red8[0];
  for (int i = 1; i < 8; ++i) m = fmaxf(m, red8[i]);
  __syncthreads();
  return m;
}

// Stage 3 h-slabs (h-1,h,h+1) x 66 w-cols x 64 z into LDS via async b128 copies.
// Out-of-range halos are loaded clamped (garbage) then zero-fixed after the wait.
__device__ __forceinline__ void stage_tile(const float* __restrict__ in,
                                           float* lds, int b, int h, int w0) {
  int t = threadIdx.x;
  unsigned lbase = (unsigned)(size_t)(void*)lds;  // low 32 bits = LDS byte offset
  unsigned long long gbase = (unsigned long long)(size_t)in;
  for (int cid = t; cid < CHUNKS; cid += 256) {
    int slab = cid / (NCOL * 16);
    int r    = cid - slab * (NCOL * 16);     // chunk within slab
    int col  = r >> 4;
    int zc   = r & 15;
    int gh = h - 1 + slab; gh = gh < 0 ? 0 : (gh > HH - 1 ? HH - 1 : gh);
    int gw = w0 - 1 + col; gw = gw < 0 ? 0 : (gw > WW - 1 ? WW - 1 : gw);
    unsigned long long ga = gbase +
        (unsigned long long)(((b * HH + gh) * WW + gw)) * (ZZ * 4) +
        (unsigned long long)(zc * 16);
    async_load_b128(lbase + (unsigned)(slab * SLAB_B + r * 16), ga);
  }
  wait_async0();
  __syncthreads();
  // Zero-fix domain boundaries.
  if (h == 0)      for (int i = t; i < SLAB; i += 256) lds[i] = 0.f;
  if (h == HH - 1) for (int i = t; i < SLAB; i += 256) lds[2 * SLAB + i] = 0.f;
  if (w0 == 0 && t < 3 * ZZ)       lds[(t / ZZ) * SLAB + (t % ZZ)] = 0.f;
  if (w0 + TW == WW && t < 3 * ZZ) lds[(t / ZZ) * SLAB + (NCOL - 1) * ZZ + (t % ZZ)] = 0.f;
  __syncthreads();
}

// Stencil on a staged float4 (z-quad). Returns out quad; cc = center quad.
__device__ __forceinline__ float4 stencil4(const float* lds, int wl, int zq, float4& cc) {
  int z4 = zq * 4;
  int idxC = SLAB + (wl + 1) * ZZ + z4;             // slab1 (h), col wl+1
  cc        = *(const float4*)&lds[idxC];
  float lft = zq ? lds[idxC - 1] : 0.f;             // z-1 across quad edge
  float rgt = (zq == 15) ? 0.f : lds[idxC + 4];     // z+1 across quad edge
  float4 wm = *(const float4*)&lds[SLAB + wl * ZZ + z4];            // w-1
  float4 wp = *(const float4*)&lds[SLAB + (wl + 2) * ZZ + z4];      // w+1
  float4 hm = *(const float4*)&lds[(wl + 1) * ZZ + z4];             // h-1
  float4 hp = *(const float4*)&lds[2 * SLAB + (wl + 1) * ZZ + z4];  // h+1
  float4 o;
  o.x = 6.f*cc.x - wm.x - 2.f*wp.x - 4.f*hm.x - 3.f*hp.x - 5.f*lft  - 6.f*cc.y;
  o.y = 6.f*cc.y - wm.y - 2.f*wp.y - 4.f*hm.y - 3.f*hp.y - 5.f*cc.x - 6.f*cc.z;
  o.z = 6.f*cc.z - wm.z - 2.f*wp.z - 4.f*hm.z - 3.f*hp.z - 5.f*cc.y - 6.f*cc.w;
  o.w = 6.f*cc.w - wm.w - 2.f*wp.w - 4.f*hm.w - 3.f*hp.w - 5.f*cc.z - 6.f*rgt;
  return o;
}

// Final fused pass: y = stencil(u4); dot += u4.y ; x = c4*u4 -> out; diff stats vs ref.
// (Defined first so the async staging path leads the emitted .text.)
__global__ __launch_bounds__(256) void k_final(const float* __restrict__ u4,
                                               const float* __restrict__ ref,
                                               const float* __restrict__ cb,
                                               float* __restrict__ xout,
                                               float* __restrict__ pDot,
                                               float* __restrict__ pLoss,
                                               float* __restrict__ pMax) {
  __shared__ __align__(16) float lds[3 * SLAB];
  __shared__ float red[8];
  int t = threadIdx.x;
  int wt = blockIdx.x, h = blockIdx.y, b = blockIdx.z;
  int w0 = wt * TW;
  stage_tile(u4, lds, b, h, w0);

  float scale = cb[b];
  float dsum = 0.f, lsum = 0.f, mx = 0.f;
#pragma unroll
  for (int i = 0; i < 4; ++i) {
    int e4 = i * 256 + t;
    int wl = e4 >> 4, zq = e4 & 15;
    float4 cc;
    float4 o = stencil4(lds, wl, zq, cc);
    dsum += cc.x*o.x + cc.y*o.y + cc.z*o.z + cc.w*o.w;   // u4 . u5 (unscaled)
    int gidx = ((b * HH + h) * WW + (w0 + wl)) * ZZ + zq * 4;
    float4 rv = *(const float4*)&ref[gidx];
    float x0 = scale*cc.x, x1 = scale*cc.y, x2 = scale*cc.z, x3 = scale*cc.w;
    // xout = d_out+1 is only 4B-aligned -> scalar stores.
    xout[gidx+0] = x0; xout[gidx+1] = x1; xout[gidx+2] = x2; xout[gidx+3] = x3;
    float d0 = x0-rv.x, d1 = x1-rv.y, d2 = x2-rv.z, d3 = x3-rv.w;
    lsum += d0*d0 + d1*d1 + d2*d2 + d3*d3;
    mx = fmaxf(mx, fmaxf(fmaxf(fabsf(d0), fabsf(d1)), fmaxf(fabsf(d2), fabsf(d3))));
  }
  int p = ((b * HH + h) << 2) + wt;
  float s;
  s = blockSum(dsum, red); if (t == 0) pDot[p]  = s;
  s = blockSum(lsum, red); if (t == 0) pLoss[p] = s;
  s = blockMax(mx,   red); if (t == 0) pMax[p]  = s;
}

// u_out = stencil(u_in); per-block partials of sumsq(u_out) (and sumsq(u_in) on 1st pass).
template <bool FIRST>
__global__ __launch_bounds__(256) void k_stencil(const float* __restrict__ in,
                                                 float* __restrict__ out,
                                                 float* __restrict__ pOut,
                                                 float* __restrict__ pIn) {
  __shared__ __align__(16) float lds[3 * SLAB];
  __shared__ float red[8];
  int t = threadIdx.x;
  int wt = blockIdx.x, h = blockIdx.y, b = blockIdx.z;
  int w0 = wt * TW;
  stage_tile(in, lds, b, h, w0);

  float lsum = 0.f, isum = 0.f;
#pragma unroll
  for (int i = 0; i < 4; ++i) {
    int e4 = i * 256 + t;               // 1024 float4s per tile
    int wl = e4 >> 4, zq = e4 & 15;
    float4 cc;
    float4 o = stencil4(lds, wl, zq, cc);
    int gidx = ((b * HH + h) * WW + (w0 + wl)) * ZZ + zq * 4;
    *(float4*)&out[gidx] = o;
    lsum += o.x*o.x + o.y*o.y + o.z*o.z + o.w*o.w;
    if (FIRST) isum += cc.x*cc.x + cc.y*cc.y + cc.z*cc.z + cc.w*cc.w;
  }
  int p = ((b * HH + h) << 2) + wt;
  float s = blockSum(lsum, red);
  if (t == 0) pOut[p] = s;
  if (FIRST) { float s2 = blockSum(isum, red); if (t == 0) pIn[p] = s2; }
}

// Reduce 1024 per-batch partials, update scale chain c[b] = prev/(prev*norm + eps).
__global__ __launch_bounds__(256) void k_rs(const float* __restrict__ part,
                                            float* __restrict__ c, int initFlag) {
  __shared__ float red[8];
  int b = blockIdx.x, t = threadIdx.x;
  float s = 0.f;
  for (int i = t; i < PPB; i += 256) s += part[b * PPB + i];
  s = blockSum(s, red);
  if (t == 0) {
    float prev = initFlag ? 1.0f : c[b];
    c[b] = prev / (prev * sqrtf(s) + 1e-6f);
  }
}

// Scalars: loss, max_error, eigenvalue = (1/B) sum_b c4[b]^2 * dot_b.
__global__ __launch_bounds__(256) void k_finalize(const float* __restrict__ pDot,
                                                  const float* __restrict__ pLoss,
                                                  const float* __restrict__ pMax,
                                                  const float* __restrict__ c,
                                                  float* __restrict__ out) {
  __shared__ float red[8];
  int t = threadIdx.x;
  float ls = 0.f, es = 0.f, mx = 0.f;
  for (int i = t; i < NBLK; i += 256) {
    ls += pLoss[i];
    mx = fmaxf(mx, pMax[i]);
    float cv = c[i >> 10];          // 1024 partials per batch, batch-major
    es += cv * cv * pDot[i];
  }
  ls = blockSum(ls, red);
  es = blockSum(es, red);
  mx = blockMax(mx, red);
  if (t == 0) {
    out[0]        = ls / (float)NTOT;  // loss
    out[NTOT + 1] = mx;                // max_error
    out[NTOT + 2] = es / (float)BB;    // eigenvalue
  }
}

extern "C" void kernel_launch(void* const* d_in, const int* in_sizes, int n_in,
                              void* d_out, int out_size, void* d_ws, size_t ws_size,
                              hipStream_t stream) {
  (void)in_sizes; (void)n_in; (void)out_size; (void)ws_size;
  const float* x   = (const float*)d_in[0];
  const float* ref = (const float*)d_in[1];
  float* outf = (float*)d_out;
  float* wsf  = (float*)d_ws;
  // ws layout (floats): [0..8) scale chain c | 32: pA[8192] | 8224: pB[8192]
  //                     | 16416: pC[8192] | 32768: bufA[N]
  float* c    = wsf;
  float* pA   = wsf + 32;
  float* pB   = wsf + 32 + NBLK;
  float* pC   = wsf + 32 + 2 * NBLK;
  float* bufA = wsf + 32768;
  float* ping = outf;       // u1,u3 live in d_out[0..N) (16B-aligned base)
  float* xout = outf + 1;   // final x region

  dim3 grid(WW / TW, HH, BB), blk(256);
  // u1 = stencil(u0); also sumsq(u0) partials
  k_stencil<true ><<<grid, blk, 0, stream>>>(x, ping, pA, pB);
  k_rs<<<BB, 256, 0, stream>>>(pB, c, 1);   // c0 from ||u0||
  k_rs<<<BB, 256, 0, stream>>>(pA, c, 0);   // c1 from ||u1||
  k_stencil<false><<<grid, blk, 0, stream>>>(ping, bufA, pA, nullptr); // u2
  k_rs<<<BB, 256, 0, stream>>>(pA, c, 0);   // c2
  k_stencil<false><<<grid, blk, 0, stream>>>(bufA, ping, pA, nullptr); // u3
  k_rs<<<BB, 256, 0, stream>>>(pA, c, 0);   // c3
  k_stencil<false><<<grid, blk, 0, stream>>>(ping, bufA, pA, nullptr); // u4
  k_rs<<<BB, 256, 0, stream>>>(pA, c, 0);   // c4
  // Fused: y=stencil(u4), dot(u4,y), x=c4*u4, loss/max vs ref
  k_final<<<grid, blk, 0, stream>>>(bufA, ref, c, xout, pA, pB, pC);
  k_finalize<<<1, 256, 0, stream>>>(pA, pB, pC, c, outf);
}